// RandomSelection_63720134803890
// MI455X (gfx1250) — compile-verified
//
#include <hip/hip_runtime.h>
#include <stdint.h>

// ---------------- problem constants (from reference) ----------------
#define NUM_GRAPHS       512
#define NODES_PER_GRAPH  1024
#define K_SEL            512                         // ceil(1024 * 0.5)
#define N_NODES          (NUM_GRAPHS * NODES_PER_GRAPH)  // 524288
#define K_TOTAL          (NUM_GRAPHS * K_SEL)            // 262144
#define C_FEAT           64
#define E_EDGES          (16 * N_NODES)                  // 8388608

// real vector types (clang ext vectors) so nontemporal builtins accept them
typedef long long v2i64 __attribute__((ext_vector_type(2)));
typedef float     v4f32 __attribute__((ext_vector_type(4)));
typedef int       v4i32 __attribute__((ext_vector_type(4)));

// ---------------- deterministic per-slot selection ----------------
__device__ __forceinline__ uint32_t mix32(uint32_t x) {
    x ^= x >> 16; x *= 0x7feb352du;
    x ^= x >> 15; x *= 0x846ca68bu;
    x ^= x >> 16;
    return x;
}
// Graph g, slot i in [0, K_SEL): kept local node id = 2*i + bit.
// Produces exactly K_SEL sorted pseudo-random local ids per graph.
__device__ __forceinline__ int sel_local(int g, int i) {
    uint32_t h = mix32(((uint32_t)(g * K_SEL + i)) ^ 0x2545F491u);
    return 2 * i + (int)(h & 1u);
}

// ---------------- CDNA5 async global<->LDS helpers ----------------
__device__ __forceinline__ void async_load_b128(uint32_t lds_off, const void* gaddr) {
    asm volatile("global_load_async_to_lds_b128 %0, %1, off"
                 :: "v"(lds_off), "v"(gaddr) : "memory");
}
__device__ __forceinline__ void async_store_b128(void* gaddr, uint32_t lds_off) {
    asm volatile("global_store_async_from_lds_b128 %0, %1, off"
                 :: "v"(gaddr), "v"(lds_off) : "memory");
}
__device__ __forceinline__ void wait_async0() {
    asm volatile("s_wait_asynccnt 0" ::: "memory");
}

// ---------------- kernel 1: relabel table init (mask = -1) ----------------
__global__ void k_mask_init(int* __restrict__ mask) {
    int t = blockIdx.x * blockDim.x + threadIdx.x;      // N_NODES/4 threads
    v4i32 m1 = {-1, -1, -1, -1};
    ((v4i32*)mask)[t] = m1;
}

// ---------------- kernel 2: selection scatter + batch_sel + perm ----------------
__global__ void k_select(int* __restrict__ mask,
                         float* __restrict__ out_batch,
                         float* __restrict__ out_perm) {
    int kpos = blockIdx.x * blockDim.x + threadIdx.x;   // [0, K_TOTAL)
    int g = kpos >> 9;                                  // /K_SEL
    int i = kpos & (K_SEL - 1);
    int local     = sel_local(g, i);
    int global_id = (g << 10) + local;                  // g*NODES_PER_GRAPH + local
    mask[global_id]  = kpos;
    out_batch[kpos]  = (float)g;
    out_perm[kpos]   = (float)global_id;                // < 2^20, exact in f32
}

// ---------------- kernel 3: x row-gather via async DMA through LDS ----------------
// 16B chunk per (thread, slot); 4 chunks per 256B row; adjacent lanes cover
// adjacent chunks so a wave's dst store is 512B: 64B in each of 8 rows. Phase-split
// pipeline: 4 async loads in flight per thread, single ASYNCcnt wait, then
// 4 async stores. Data never touches VGPRs.
__global__ void k_gather_async(const float* __restrict__ x, float* __restrict__ out_x) {
    __shared__ __align__(16) unsigned char smem[256 * 64];   // 4 x 16B slots / thread
    const int tid = threadIdx.x;
    const uint32_t slot_base = (uint32_t)(size_t)(&smem[tid * 64]);

    const int stride = gridDim.x * blockDim.x;               // 1048576 threads
    const int chunk0 = blockIdx.x * blockDim.x + tid;        // 4194304 chunks total

    void* dst[4];
#pragma unroll
    for (int it = 0; it < 4; ++it) {
        int chunk = chunk0 + it * stride;
        int kpos  = chunk >> 2;
        int q     = chunk & 3;
        int g = kpos >> 9;
        int i = kpos & (K_SEL - 1);
        int local = sel_local(g, i);
        const void* src = (const void*)(x + ((size_t)((g << 10) + local)) * C_FEAT + q * 4);
        dst[it]         = (void*)(out_x + (size_t)kpos * C_FEAT + q * 4);
        async_load_b128(slot_base + (uint32_t)(it * 16), src);   // HBM -> LDS
    }
    wait_async0();                                               // all 4 landed
#pragma unroll
    for (int it = 0; it < 4; ++it) {
        async_store_b128(dst[it], slot_base + (uint32_t)(it * 16));  // LDS -> HBM
    }
    wait_async0();
}

// ---------------- kernel 4: edge filter / relabel ----------------
// 4 edges per thread: B128 NT loads on the ~200MB edge stream (keeps the hot
// 2MB mask table resident in the 192MB L2), 8 mask lookups, B128 NT stores.
__global__ void k_edges(const long long* __restrict__ ei,
                        const int* __restrict__ mask,
                        float* __restrict__ out_ei) {
    const long long* row0 = ei;
    const long long* row1 = ei + (size_t)E_EDGES;
    int t = blockIdx.x * blockDim.x + threadIdx.x;      // E/4 threads
    size_t e = (size_t)t * 4;

    v2i64 s01 = __builtin_nontemporal_load((const v2i64*)(row0 + e));
    v2i64 s23 = __builtin_nontemporal_load((const v2i64*)(row0 + e + 2));
    v2i64 d01 = __builtin_nontemporal_load((const v2i64*)(row1 + e));
    v2i64 d23 = __builtin_nontemporal_load((const v2i64*)(row1 + e + 2));

    int r0 = mask[(int)s01.x], r1 = mask[(int)s01.y];
    int r2 = mask[(int)s23.x], r3 = mask[(int)s23.y];
    int c0 = mask[(int)d01.x], c1 = mask[(int)d01.y];
    int c2 = mask[(int)d23.x], c3 = mask[(int)d23.y];

    bool k0 = (r0 >= 0) & (c0 >= 0);
    bool k1 = (r1 >= 0) & (c1 >= 0);
    bool k2 = (r2 >= 0) & (c2 >= 0);
    bool k3 = (r3 >= 0) & (c3 >= 0);

    v4f32 orow = {k0 ? (float)r0 : -1.0f, k1 ? (float)r1 : -1.0f,
                  k2 ? (float)r2 : -1.0f, k3 ? (float)r3 : -1.0f};
    v4f32 ocol = {k0 ? (float)c0 : -1.0f, k1 ? (float)c1 : -1.0f,
                  k2 ? (float)c2 : -1.0f, k3 ? (float)c3 : -1.0f};

    __builtin_nontemporal_store(orow, (v4f32*)(out_ei + e));
    __builtin_nontemporal_store(ocol, (v4f32*)(out_ei + (size_t)E_EDGES + e));
}

// ---------------- launcher ----------------
extern "C" void kernel_launch(void* const* d_in, const int* in_sizes, int n_in,
                              void* d_out, int out_size, void* d_ws, size_t ws_size,
                              hipStream_t stream) {
    const float*     x  = (const float*)d_in[0];       // [N, 64] f32
    const long long* ei = (const long long*)d_in[1];   // [2, E] i64

    float* out       = (float*)d_out;
    float* out_x     = out;                                      // K*64
    float* out_ei    = out_x  + (size_t)K_TOTAL * C_FEAT;        // 2*E
    float* out_batch = out_ei + (size_t)2 * E_EDGES;             // K
    float* out_perm  = out_batch + (size_t)K_TOTAL;              // K

    int* mask = (int*)d_ws;                                      // N * int32 = 2 MB

    k_mask_init   <<<N_NODES / 4 / 256, 256, 0, stream>>>(mask);
    k_select      <<<K_TOTAL / 256,     256, 0, stream>>>(mask, out_batch, out_perm);
    k_gather_async<<<4096,              256, 0, stream>>>(x, out_x);
    k_edges       <<<(E_EDGES / 4)/256, 256, 0, stream>>>(ei, mask, out_ei);
}